// STSLateFusion2Inv_2138893713636
// MI455X (gfx1250) — compile-verified
//
#include <hip/hip_runtime.h>
#include <stdint.h>

// ---------------------------------------------------------------------------
// Cost-volume construction (stereo). Pure data movement: ~755 MB of output
// stores per call -> HBM-store-bandwidth bound (~32 us floor @ 23.3 TB/s).
// gfx1250 paths used: TDM tensor_load_to_lds (TENSORcnt) to stage each input
// row in LDS, ds_load_b32 gathers for the shifted right-image slices, and
// non-temporal global_store_b128 for the huge streaming output.
// ---------------------------------------------------------------------------

typedef __attribute__((ext_vector_type(4))) float        v4f;
typedef __attribute__((ext_vector_type(4))) unsigned int v4u;
typedef __attribute__((ext_vector_type(8))) int          v8i;
typedef __attribute__((ext_vector_type(4))) int          v4i;

#define CC 32   // channels per image
#define HH 96
#define WW 240
#define DD 64   // max_disp(192) / 3

__global__ __launch_bounds__(256) void STSLateFusion2Inv_cost_volume_kernel(
    const float* __restrict__ l0, const float* __restrict__ r0,
    const float* __restrict__ l1, const float* __restrict__ r1,
    float* __restrict__ out)
{
    __shared__ float row[256];   // 240 used (one W-row), 1 KB LDS

    const int bid = blockIdx.x;            // bid = (s*2C + c2)*H + h
    const int h   = bid % HH;
    const int c2  = (bid / HH) % (2 * CC);
    const int s   = bid / (HH * 2 * CC);

    const bool isRight = (c2 >= CC);
    const int  c       = isRight ? (c2 - CC) : c2;
    const float* img   = s ? (isRight ? r1 : l1) : (isRight ? r0 : l0);
    const float* grow  = img + ((size_t)c * HH + h) * WW;   // source row

    const int t = threadIdx.x;

    // ---- Stage the 240-float row into LDS via the Tensor Data Mover ----
    // Minimal D#: 1 descriptor, 4-byte elements, 1 row of tile_dim0=240.
    if (t < 32) {   // one wave issues the DMA (EXEC ignored by TDM)
        const uint32_t lds_off = (uint32_t)(uintptr_t)(&row[0]); // low 32 bits of flat LDS addr = LDS byte offset
        const uint64_t gaddr   = (uint64_t)(uintptr_t)grow;

        v4u g0;
        g0[0] = 1u;                                        // count=1, user descriptor, gather off
        g0[1] = lds_off;                                   // lds_addr (bytes)
        g0[2] = (uint32_t)(gaddr & 0xFFFFFFFFu);           // global_addr[31:0]
        g0[3] = (uint32_t)((gaddr >> 32) & 0x01FFFFFFu)    // global_addr[56:32]
              | (2u << 30);                                // type = 2 ("image")

        v8i g1;
        g1[0] = (int)(2u << 16);                           // data_size = 2 -> 4 bytes
        g1[1] = (int)((uint32_t)WW << 16);                 // tensor_dim0 = 240 (bits 79:48 low half)
        g1[2] = (int)(1u << 16);                           // tensor_dim1 = 1  (bits 111:80 low half)
        g1[3] = (int)((uint32_t)WW << 16);                 // tile_dim0 = 240  (bits 127:112)
        g1[4] = 1;                                         // tile_dim1 = 1, tile_dim2 = 0
        g1[5] = WW;                                        // tensor_dim0_stride = 240
        g1[6] = 0;                                         // stride hi / tensor_dim1_stride lo
        g1[7] = 0;                                         // tensor_dim1_stride hi

        v4i zz4 = {0, 0, 0, 0};                            // groups 2/3 unused (<=2D tensor)
        v8i zz8 = {0, 0, 0, 0, 0, 0, 0, 0};                // extra group (6-arg toolchain form)
        __builtin_amdgcn_tensor_load_to_lds(g0, g1, zz4, zz4, zz8, 0);
        __builtin_amdgcn_s_wait_tensorcnt(0);
    }
    __syncthreads();                                       // LDS row valid for all 8 waves

    // ---- Emit the 64(d) x 240(w) tile with NT b128 stores ----
    const int q     = t & 63;          // float4 chunk within the row (0..59 active)
    const int dbase = t >> 6;          // d-stripe 0..3
    if (q >= WW / 4) return;
    const int w0 = q * 4;

    float* obase = out
        + ((((size_t)s * (2 * CC) + c2) * DD) * HH + h) * (size_t)WW + w0;
    const size_t dstride = (size_t)HH * WW;                // 23040 floats per d-slice

    if (!isRight) {
        // left: value independent of d, only the (w >= d) mask varies
        v4f v;
        v.x = row[w0 + 0]; v.y = row[w0 + 1];
        v.z = row[w0 + 2]; v.w = row[w0 + 3];
        #pragma unroll 4
        for (int d = dbase; d < DD; d += 4) {
            v4f o;
            o.x = (w0 + 0 >= d) ? v.x : 0.0f;
            o.y = (w0 + 1 >= d) ? v.y : 0.0f;
            o.z = (w0 + 2 >= d) ? v.z : 0.0f;
            o.w = (w0 + 3 >= d) ? v.w : 0.0f;
            __builtin_nontemporal_store(o, (v4f*)(obase + (size_t)d * dstride));
        }
    } else {
        // right: shifted gather row[w-d] from LDS, zero where w < d
        #pragma unroll 4
        for (int d = dbase; d < DD; d += 4) {
            const int b = w0 - d;
            v4f o;
            int i0 = b + 0; int i1 = b + 1; int i2 = b + 2; int i3 = b + 3;
            float f0 = row[i0 < 0 ? 0 : i0];
            float f1 = row[i1 < 0 ? 0 : i1];
            float f2 = row[i2 < 0 ? 0 : i2];
            float f3 = row[i3 < 0 ? 0 : i3];
            o.x = (i0 >= 0) ? f0 : 0.0f;
            o.y = (i1 >= 0) ? f1 : 0.0f;
            o.z = (i2 >= 0) ? f2 : 0.0f;
            o.w = (i3 >= 0) ? f3 : 0.0f;
            __builtin_nontemporal_store(o, (v4f*)(obase + (size_t)d * dstride));
        }
    }
}

extern "C" void kernel_launch(void* const* d_in, const int* in_sizes, int n_in,
                              void* d_out, int out_size, void* d_ws, size_t ws_size,
                              hipStream_t stream) {
    (void)in_sizes; (void)n_in; (void)out_size; (void)d_ws; (void)ws_size;
    const float* l0 = (const float*)d_in[0];
    const float* r0 = (const float*)d_in[1];
    const float* l1 = (const float*)d_in[2];
    const float* r1 = (const float*)d_in[3];
    // d_in[4] = max_disp (scalar); D = 192/3 = 64 baked into the kernel.
    float* out = (float*)d_out;

    const int blocks = 2 * (2 * CC) * HH;   // s * c2 * h = 12288
    STSLateFusion2Inv_cost_volume_kernel<<<blocks, 256, 0, stream>>>(l0, r0, l1, r1, out);
}